// Model_69011534512479
// MI455X (gfx1250) — compile-verified
//
#include <hip/hip_runtime.h>
#include <math.h>
#include <stdint.h>

#define NAGENTS 11
#define KCH 5
#define HW 4096   // 64*64

#if defined(__has_builtin)
#if __has_builtin(__builtin_amdgcn_global_load_async_to_lds_b128) && \
    __has_builtin(__builtin_amdgcn_s_wait_asynccnt)
#define USE_ASYNC_LDS 1
#endif
#endif
#ifndef USE_ASYNC_LDS
#define USE_ASYNC_LDS 0
#endif

typedef __attribute__((ext_vector_type(4))) int v4i;
typedef v4i __attribute__((address_space(1))) g_v4i;   // global
typedef v4i __attribute__((address_space(3))) l_v4i;   // LDS

__device__ __forceinline__ float block_reduce_max(float v, float* red)
{
#pragma unroll
    for (int off = 16; off > 0; off >>= 1)
        v = fmaxf(v, __shfl_down(v, off, 32));
    if ((threadIdx.x & 31) == 0) red[threadIdx.x >> 5] = v;
    __syncthreads();
    float r = red[0];
#pragma unroll
    for (int w = 1; w < 8; ++w) r = fmaxf(r, red[w]);
    __syncthreads();
    return r;
}

__device__ __forceinline__ float block_reduce_sum(float v, float* red)
{
#pragma unroll
    for (int off = 16; off > 0; off >>= 1)
        v += __shfl_down(v, off, 32);
    if ((threadIdx.x & 31) == 0) red[threadIdx.x >> 5] = v;
    __syncthreads();
    float r = red[0];
#pragma unroll
    for (int w = 1; w < 8; ++w) r += red[w];
    __syncthreads();
    return r;
}

__global__ __launch_bounds__(256) void fly_kpts_kernel(const float* __restrict__ logits,
                                                       float* __restrict__ out)
{
    __shared__ float sh_sum[HW];   // channel-summed heatmap
    __shared__ float bufA[HW];     // channel stage / later: NMS result ("without_top")
    __shared__ float bufB[HW];     // channel stage / later: row-max scratch
    __shared__ float red[8];
    __shared__ float s_max[KCH], s_sumk[KCH], s_hx2[KCH], s_hy2[KCH];
    __shared__ int   s_cy[NAGENTS], s_cx[NAGENTS];
    __shared__ float bvals[8];
    __shared__ int   bidxs[8];

    const int b = blockIdx.x;
    const int t = threadIdx.x;
    const float* gbase = logits + (size_t)b * KCH * HW;

    // zero the summed map (thread-owned float4 chunks; same owner does += later)
    {
        float4 z = make_float4(0.f, 0.f, 0.f, 0.f);
#pragma unroll
        for (int j = 0; j < 4; ++j) ((float4*)sh_sum)[t + 256 * j] = z;
    }

    // stage one channel (16 KB) into LDS: async DMA path on gfx1250, plain copy otherwise
    auto issue_ch = [&](int k, float* buf) {
        const float4* g4 = (const float4*)(gbase + (size_t)k * HW);
        float4* l4 = (float4*)buf;
#if USE_ASYNC_LDS
#pragma unroll
        for (int j = 0; j < 4; ++j) {
            __builtin_amdgcn_global_load_async_to_lds_b128(
                (g_v4i*)(uintptr_t)(g4 + t + 256 * j),
                (l_v4i*)(uint32_t)(uintptr_t)(l4 + t + 256 * j),
                0, 0);
        }
#else
#pragma unroll
        for (int j = 0; j < 4; ++j) l4[t + 256 * j] = g4[t + 256 * j];
#endif
    };

    // ---- Phase A: per-channel spatial softmax; accumulate summed, hx2, hy2 ----
    issue_ch(0, bufA);
    for (int k = 0; k < KCH; ++k) {
        float* cur = (k & 1) ? bufB : bufA;
#if USE_ASYNC_LDS
        __builtin_amdgcn_s_wait_asynccnt(0);
#endif
        __syncthreads();
        if (k + 1 < KCH) issue_ch(k + 1, (k & 1) ? bufA : bufB);  // overlap next channel

        float v[16];
#pragma unroll
        for (int j = 0; j < 4; ++j) {
            float4 q = ((float4*)cur)[t + 256 * j];
            v[4 * j + 0] = q.x; v[4 * j + 1] = q.y;
            v[4 * j + 2] = q.z; v[4 * j + 3] = q.w;
        }
        float mx = -INFINITY;
#pragma unroll
        for (int i = 0; i < 16; ++i) mx = fmaxf(mx, v[i]);
        mx = block_reduce_max(mx, red);

        float e[16];
        float s = 0.f;
#pragma unroll
        for (int i = 0; i < 16; ++i) { e[i] = expf(v[i] - mx); s += e[i]; }
        s = block_reduce_sum(s, red);

        float hx2 = 0.f, hy2 = 0.f;
#pragma unroll
        for (int j = 0; j < 4; ++j) {
            const int cidx = t + 256 * j;
#pragma unroll
            for (int ee = 0; ee < 4; ++ee) {
                const int idx = 4 * cidx + ee;
                const float h = e[4 * j + ee] / s;
                sh_sum[idx] += h;
                const float xl = (float)(idx & 63) * (2.0f / 63.0f) - 1.0f;
                const float yl = (float)(idx >> 6) * (2.0f / 63.0f) - 1.0f;
                hx2 += h * xl * xl;
                hy2 += h * yl * yl;
            }
        }
        hx2 = block_reduce_sum(hx2, red);
        hy2 = block_reduce_sum(hy2, red);
        if (t == 0) { s_max[k] = mx; s_sumk[k] = s; s_hx2[k] = hx2; s_hy2[k] = hy2; }
    }
    __syncthreads();

    // ---- Phase B: separable 15x15 NMS on sh_sum -> bufA holds nms result ----
    for (int j = 0; j < 16; ++j) {
        const int idx = t + 256 * j;
        const int y = idx >> 6, x = idx & 63;
        const int x0 = max(x - 7, 0), x1 = min(x + 7, 63);
        float m = -INFINITY;
        for (int xx = x0; xx <= x1; ++xx) m = fmaxf(m, sh_sum[(y << 6) + xx]);
        bufB[idx] = m;
    }
    __syncthreads();
    for (int j = 0; j < 16; ++j) {
        const int idx = t + 256 * j;
        const int y = idx >> 6, x = idx & 63;
        const int y0 = max(y - 7, 0), y1 = min(y + 7, 63);
        float m = -INFINITY;
        for (int yy = y0; yy <= y1; ++yy) m = fmaxf(m, bufB[(yy << 6) + x]);
        const float sv = sh_sum[idx];
        bufA[idx] = (sv == m) ? sv : 0.0f;
    }
    __syncthreads();

    // ---- Phase C: 11x (first-index argmax over 4096, then zero circle r<=10) ----
    for (int a = 0; a < NAGENTS; ++a) {
        float bv = -INFINITY;
        int bi = HW;
#pragma unroll
        for (int j = 0; j < 16; ++j) {
            const int idx = t + 256 * j;
            const float vv = bufA[idx];
            if (vv > bv || (vv == bv && idx < bi)) { bv = vv; bi = idx; }
        }
#pragma unroll
        for (int off = 16; off > 0; off >>= 1) {
            const float ov = __shfl_down(bv, off, 32);
            const int   oi = __shfl_down(bi, off, 32);
            if (ov > bv || (ov == bv && oi < bi)) { bv = ov; bi = oi; }
        }
        if ((t & 31) == 0) { bvals[t >> 5] = bv; bidxs[t >> 5] = bi; }
        __syncthreads();
        bv = bvals[0]; bi = bidxs[0];
#pragma unroll
        for (int w = 1; w < 8; ++w) {
            const float ov = bvals[w]; const int oi = bidxs[w];
            if (ov > bv || (ov == bv && oi < bi)) { bv = ov; bi = oi; }
        }
        const int cy = bi >> 6, cx = bi & 63;
        if (t == 0) { s_cy[a] = cy; s_cx[a] = cx; }
        const int y0 = max(cy - 10, 0), y1 = min(cy + 10, 63);
        const int x0 = max(cx - 10, 0), x1 = min(cx + 10, 63);
        const int BW = x1 - x0 + 1;
        const int N  = BW * (y1 - y0 + 1);
        for (int i = t; i < N; i += 256) {
            const int r = y0 + i / BW, c = x0 + i % BW;
            const int dy = r - cy, dx = c - cx;
            if (dx * dx + dy * dy <= 100) bufA[(r << 6) + c] = 0.f;
        }
        __syncthreads();
    }

    // ---- Phase D: per (agent,channel) masked moments from 21x21 box (1 combo / wave) ----
    const int wid = t >> 5, lane = t & 31;
    for (int combo = wid; combo < NAGENTS * KCH; combo += 8) {
        const int a = combo / KCH, k = combo % KCH;
        const int cy = s_cy[a], cx = s_cx[a];
        const float mx = s_max[k], sk = s_sumk[k];
        const int y0 = max(cy - 10, 0), y1 = min(cy + 10, 63);
        const int x0 = max(cx - 10, 0), x1 = min(cx + 10, 63);
        const float* gk = gbase + (size_t)k * HW;
        const int c = x0 + lane;
        const bool act = (c <= x1);
        const int dx = c - cx;
        const float xs = (float)dx * (2.0f / 63.0f);
        float S0 = 0.f, Sx = 0.f, Sy = 0.f, Sxy = 0.f, cf = 0.f;
        for (int r = y0; r <= y1; ++r) {
            const int dy = r - cy;
            if (act && (dx * dx + dy * dy <= 100)) {
                const float w = expf(gk[(r << 6) + c] - mx) / sk;  // == heatmap value
                const float ys = (float)dy * (2.0f / 63.0f);
                S0 += w; Sx += w * xs; Sy += w * ys; Sxy += w * xs * ys;
                cf = fmaxf(cf, w);
            }
        }
#pragma unroll
        for (int off = 16; off > 0; off >>= 1) {
            S0  += __shfl_down(S0, off, 32);
            Sx  += __shfl_down(Sx, off, 32);
            Sy  += __shfl_down(Sy, off, 32);
            Sxy += __shfl_down(Sxy, off, 32);
            cf = fmaxf(cf, __shfl_down(cf, off, 32));
        }
        if (lane == 0) {
            const float px = (float)cx * (2.0f / 63.0f) - 1.0f;
            const float py = (float)cy * (2.0f / 63.0f) - 1.0f;
            const float ux = px + Sx / S0;
            const float uy = py + Sy / S0;
            const float vx = fmaxf(s_hx2[k] - ux * ux, 1e-6f);
            const float vy = fmaxf(s_hy2[k] - uy * uy, 1e-6f);
            const float cov = Sxy - Sx * Sy / S0;
            float* o = out + ((size_t)b * (NAGENTS * KCH) + combo) * 6;
            o[0] = ux; o[1] = uy; o[2] = vx; o[3] = vy; o[4] = cov; o[5] = cf;
        }
    }
}

extern "C" void kernel_launch(void* const* d_in, const int* in_sizes, int n_in,
                              void* d_out, int out_size, void* d_ws, size_t ws_size,
                              hipStream_t stream)
{
    const float* logits = (const float*)d_in[0];
    float* out = (float*)d_out;
    const int B = in_sizes[0] / (KCH * HW);   // 256
    fly_kpts_kernel<<<B, 256, 0, stream>>>(logits, out);
}